// GraphEncoderNet_37340445671624
// MI455X (gfx1250) — compile-verified
//
#include <hip/hip_runtime.h>

// ---------------------------------------------------------------------------
// GAT encoder for MI455X (gfx1250): fp32 WMMA GEMM + atomic edge softmax.
// ---------------------------------------------------------------------------

typedef __attribute__((ext_vector_type(2))) float v2f;
typedef __attribute__((ext_vector_type(8))) float v8f;

#define HID 512
#define N_S_TOT 20000   // 32 * 625
#define N_V_TOT 320     // 32 * 10
#define BK 32           // K panel depth staged in LDS

// ---------------------------------------------------------------------------
// GEMM: H[N x 512] = X[N x K] @ W[K x 512]  using V_WMMA_F32_16X16X4_F32.
// Block = 256 threads (8 waves). Block tile 64x128; each wave owns 16x64
// (4 accumulators), reusing one A fragment across 4 WMMAs.
// W is staged into LDS pre-swizzled as Ws4[kq][n][half][j] (k = 4kq+2h+j) so
// each lane's B fragment is one aligned, bank-conflict-free ds_load_b64.
// ---------------------------------------------------------------------------
__global__ __launch_bounds__(256)
void gat_gemm_wmma(const float* __restrict__ X, const float* __restrict__ W,
                   float* __restrict__ H, int Nrows, int K) {
  __shared__ float Xs[64][36];          // 64 rows x 32 k, +4 pad
  __shared__ float Ws4[8 * 128 * 4];    // swizzled 32k x 128n panel (16 KB)

  const int tid  = threadIdx.x;
  const int lane = tid & 31;
  const int wave = tid >> 5;
  const int wr   = wave >> 1;           // wave row-tile 0..3 (16 rows each)
  const int wc   = wave & 1;            // wave col-tile 0..1 (64 cols each)
  const int m    = lane & 15;           // M (A) / N (B,C) index within tile
  const int half = lane >> 4;           // half-wave: selects K pair / +8 rows
  const int kh   = half << 1;           // K sub-offset: 0 or 2

  const int row0 = blockIdx.x * 64;
  const int col0 = blockIdx.y * 128;

  v8f acc[4] = {v8f{}, v8f{}, v8f{}, v8f{}};

  for (int k0 = 0; k0 < K; k0 += BK) {
    // ---- stage X panel (64x32): 512 float4, 2 per thread ----
#pragma unroll
    for (int i = 0; i < 2; ++i) {
      int slot = tid + 256 * i;          // 0..511
      int r    = slot >> 3;
      int c4   = (slot & 7) << 2;
      float4 v = make_float4(0.f, 0.f, 0.f, 0.f);
      int gr = row0 + r;
      if (gr < Nrows) v = *(const float4*)(X + (size_t)gr * K + k0 + c4);
      *(float4*)&Xs[r][c4] = v;
    }
    // ---- stage W panel (32x128) swizzled: 1024 float4, 4 per thread ----
#pragma unroll
    for (int i = 0; i < 4; ++i) {
      int slot = tid + 256 * i;          // 0..1023
      int r    = slot >> 5;              // k row 0..31
      int c4   = (slot & 31) << 2;       // col 0..124
      float4 v = *(const float4*)(W + (size_t)(k0 + r) * HID + col0 + c4);
      int base = ((r >> 2) << 9) + (((r >> 1) & 1) << 1) + (r & 1);
      Ws4[base + ((c4 + 0) << 2)] = v.x;
      Ws4[base + ((c4 + 1) << 2)] = v.y;
      Ws4[base + ((c4 + 2) << 2)] = v.z;
      Ws4[base + ((c4 + 3) << 2)] = v.w;
    }
    // prefetch next W panel (global_prefetch_b8)
    if (k0 + BK < K)
      __builtin_prefetch(W + (size_t)(k0 + BK + (tid >> 3)) * HID + col0, 0, 3);
    __syncthreads();

    // ---- 8 k-steps x 4 WMMA over the 32-deep panel ----
#pragma unroll
    for (int kk = 0; kk < BK; kk += 4) {
      v2f A = *(const v2f*)&Xs[wr * 16 + m][kk + kh];
      const float* bbase = &Ws4[((kk >> 2) << 9) + ((wc * 64 + m) << 2) + kh];
#pragma unroll
      for (int t = 0; t < 4; ++t) {
        v2f B = *(const v2f*)(bbase + (t << 6));   // t*16 cols * 4
        acc[t] = __builtin_amdgcn_wmma_f32_16x16x4_f32(
            false, A, false, B, (short)0, acc[t], false, false);
      }
    }
    __syncthreads();
  }

  // ---- store C/D: VGPR r -> row (r + 8*half), lane%16 -> col ----
#pragma unroll
  for (int t = 0; t < 4; ++t) {
#pragma unroll
    for (int r = 0; r < 8; ++r) {
      int grow = row0 + wr * 16 + r + 8 * half;
      if (grow < Nrows)
        H[(size_t)grow * HID + col0 + wc * 64 + t * 16 + m] = acc[t][r];
    }
  }
}

// ---------------------------------------------------------------------------
// el = h . al, er = h . ar  (one wave32 per node row)
// ---------------------------------------------------------------------------
__global__ __launch_bounds__(256)
void proj_kernel(const float* __restrict__ h, const float* __restrict__ al,
                 const float* __restrict__ ar, float* __restrict__ el,
                 float* __restrict__ er, int N) {
  int lane = threadIdx.x & 31;
  int row  = blockIdx.x * 8 + (threadIdx.x >> 5);
  if (row >= N) return;
  const float* hr = h + (size_t)row * HID;
  float sl = 0.f, sr = 0.f;
#pragma unroll
  for (int j = 0; j < HID / 32; ++j) {
    int c = lane + 32 * j;
    float v = hr[c];
    sl += v * al[c];
    sr += v * ar[c];
  }
#pragma unroll
  for (int off = 16; off > 0; off >>= 1) {
    sl += __shfl_xor(sl, off, 32);
    sr += __shfl_xor(sr, off, 32);
  }
  if (lane == 0) { el[row] = sl; er[row] = sr; }
}

__global__ void fill_f32(float* __restrict__ p, float v, size_t n) {
  size_t i = (size_t)blockIdx.x * blockDim.x + threadIdx.x;
  if (i < n) p[i] = v;
}

__global__ void edge_score_kernel(const float* __restrict__ el,
                                  const float* __restrict__ er,
                                  const int* __restrict__ src,
                                  const int* __restrict__ dst,
                                  float* __restrict__ e, int E) {
  int i = blockIdx.x * blockDim.x + threadIdx.x;
  if (i >= E) return;
  float v = el[src[i]] + er[dst[i]];
  e[i] = v > 0.f ? v : 0.2f * v;   // LeakyReLU(0.2)
}

__device__ __forceinline__ void atomicMaxF(float* addr, float val) {
  if (val >= 0.f) atomicMax((int*)addr, __float_as_int(val));
  else            atomicMin((unsigned int*)addr, (unsigned int)__float_as_int(val));
}

__global__ void seg_max_kernel(const float* __restrict__ e,
                               const int* __restrict__ dst,
                               float* __restrict__ m, int E) {
  int i = blockIdx.x * blockDim.x + threadIdx.x;
  if (i >= E) return;
  atomicMaxF(&m[dst[i]], e[i]);
}

__global__ void edge_p_kernel(const float* __restrict__ e,
                              const float* __restrict__ m,
                              const int* __restrict__ dst,
                              float* __restrict__ p,
                              float* __restrict__ denom, int E) {
  int i = blockIdx.x * blockDim.x + threadIdx.x;
  if (i >= E) return;
  int d = dst[i];
  float mm = m[d];
  if (!(mm >= -3.0e38f)) mm = 0.f;     // zero in-degree guard (-inf -> 0)
  float pv = expf(e[i] - mm);
  p[i] = pv;
  atomicAdd(&denom[d], pv);
}

// one wave32 per edge: out[dst] += (p/denom[dst]) * h[src]
__global__ __launch_bounds__(256)
void aggregate_kernel(const float* __restrict__ p, const float* __restrict__ denom,
                      const float* __restrict__ h, const int* __restrict__ src,
                      const int* __restrict__ dst, float* __restrict__ out, int E) {
  int lane = threadIdx.x & 31;
  int eidx = blockIdx.x * 8 + (threadIdx.x >> 5);
  if (eidx >= E) return;
  int s = src[eidx], d = dst[eidx];
  float coef = p[eidx] / denom[d];
  const float* hs = h + (size_t)s * HID;
  float* od = out + (size_t)d * HID;
#pragma unroll
  for (int j = 0; j < HID / 32; ++j) {
    int c = lane + 32 * j;
    atomicAdd(&od[c], coef * hs[c]);
  }
}

__global__ void bias_kernel(float* __restrict__ out, const float* __restrict__ b,
                            size_t total) {
  size_t i = (size_t)blockIdx.x * blockDim.x + threadIdx.x;
  if (i < total) out[i] += b[i & (HID - 1)];
}

// out[b][h][n] = nf[b*Nn + n][h]   (swapaxes(1,2))
__global__ void transpose_out_kernel(const float* __restrict__ nf,
                                     float* __restrict__ out, int Nn, long total) {
  long idx = (long)blockIdx.x * blockDim.x + threadIdx.x;
  if (idx >= total) return;
  int  n = (int)(idx % Nn);
  long t = idx / Nn;
  int  h = (int)(t % HID);
  int  b = (int)(t / HID);
  out[idx] = nf[((long)b * Nn + n) * HID + h];
}

// ---------------------------------------------------------------------------
// Host-side layer driver (all launches on `stream`)
// ---------------------------------------------------------------------------
static void launch_gat_layer(const float* x, const float* W, const float* al,
                             const float* ar, const float* b, const int* src,
                             const int* dst, float* h, float* out, float* el,
                             float* er, float* m, float* denom, float* e,
                             float* p, int N, int E, int K, hipStream_t stream) {
  dim3 gg((unsigned)((N + 63) / 64), HID / 128);
  gat_gemm_wmma<<<gg, 256, 0, stream>>>(x, W, h, N, K);
  proj_kernel<<<(N + 7) / 8, 256, 0, stream>>>(h, al, ar, el, er, N);
  edge_score_kernel<<<(E + 255) / 256, 256, 0, stream>>>(el, er, src, dst, e, E);
  fill_f32<<<(N + 255) / 256, 256, 0, stream>>>(m, -__builtin_huge_valf(), (size_t)N);
  seg_max_kernel<<<(E + 255) / 256, 256, 0, stream>>>(e, dst, m, E);
  fill_f32<<<(N + 255) / 256, 256, 0, stream>>>(denom, 0.f, (size_t)N);
  edge_p_kernel<<<(E + 255) / 256, 256, 0, stream>>>(e, m, dst, p, denom, E);
  size_t tot = (size_t)N * HID;
  fill_f32<<<(unsigned)((tot + 255) / 256), 256, 0, stream>>>(out, 0.f, tot);
  aggregate_kernel<<<(E + 7) / 8, 256, 0, stream>>>(p, denom, h, src, dst, out, E);
  bias_kernel<<<(unsigned)((tot + 255) / 256), 256, 0, stream>>>(out, b, tot);
}

extern "C" void kernel_launch(void* const* d_in, const int* in_sizes, int n_in,
                              void* d_out, int out_size, void* d_ws, size_t ws_size,
                              hipStream_t stream) {
  (void)n_in; (void)out_size; (void)ws_size;
  const float* s_x = (const float*)d_in[0];
  const float* v_x = (const float*)d_in[1];
  const float *sW[3], *sal[3], *sar[3], *sb[3];
  const float *vW[3], *val_[3], *var_[3], *vb[3];
  for (int l = 0; l < 3; ++l) {
    sW[l]   = (const float*)d_in[2 + 4 * l + 0];
    sal[l]  = (const float*)d_in[2 + 4 * l + 1];
    sar[l]  = (const float*)d_in[2 + 4 * l + 2];
    sb[l]   = (const float*)d_in[2 + 4 * l + 3];
    vW[l]   = (const float*)d_in[14 + 4 * l + 0];
    val_[l] = (const float*)d_in[14 + 4 * l + 1];
    var_[l] = (const float*)d_in[14 + 4 * l + 2];
    vb[l]   = (const float*)d_in[14 + 4 * l + 3];
  }
  const int* ssrc = (const int*)d_in[26];
  const int* sdst = (const int*)d_in[27];
  const int* vsrc = (const int*)d_in[28];
  const int* vdst = (const int*)d_in[29];
  const int Es = in_sizes[26];   // 320000
  const int Ev = in_sizes[28];   // 1280

  // ---- workspace layout (floats) ----
  float* ws = (float*)d_ws;
  size_t o = 0;
  float* sA  = ws + o; o += (size_t)N_S_TOT * HID;
  float* sH  = ws + o; o += (size_t)N_S_TOT * HID;
  float* sel = ws + o; o += N_S_TOT;
  float* ser = ws + o; o += N_S_TOT;
  float* sm  = ws + o; o += N_S_TOT;
  float* sdn = ws + o; o += N_S_TOT;
  float* se  = ws + o; o += (size_t)Es;
  float* sp  = ws + o; o += (size_t)Es;
  float* vA  = ws + o; o += (size_t)N_V_TOT * HID;
  float* vH  = ws + o; o += (size_t)N_V_TOT * HID;
  float* vel = ws + o; o += N_V_TOT;
  float* ver = ws + o; o += N_V_TOT;
  float* vm  = ws + o; o += N_V_TOT;
  float* vdn = ws + o; o += N_V_TOT;
  float* ve  = ws + o; o += (size_t)Ev;
  float* vp  = ws + o; o += (size_t)Ev;

  // ---- static encoder: K = 256, 512, 512; x dead after GEMM -> out reuses sA
  const int sK[3] = {256, HID, HID};
  const float* x = s_x;
  for (int l = 0; l < 3; ++l) {
    launch_gat_layer(x, sW[l], sal[l], sar[l], sb[l], ssrc, sdst,
                     sH, sA, sel, ser, sm, sdn, se, sp,
                     N_S_TOT, Es, sK[l], stream);
    x = sA;
  }

  // ---- vnr encoder: K = 128, 512, 512
  const int vK[3] = {128, HID, HID};
  x = v_x;
  for (int l = 0; l < 3; ++l) {
    launch_gat_layer(x, vW[l], val_[l], var_[l], vb[l], vsrc, vdst,
                     vH, vA, vel, ver, vm, vdn, ve, vp,
                     N_V_TOT, Ev, vK[l], stream);
    x = vA;
  }

  // ---- outputs: (vnr_hidden [32,512,10], static_hidden [32,512,625]) ----
  float* outp = (float*)d_out;
  long vtot = (long)32 * HID * 10;
  long stot = (long)32 * HID * 625;
  transpose_out_kernel<<<(unsigned)((vtot + 255) / 256), 256, 0, stream>>>(
      vA, outp, 10, vtot);
  transpose_out_kernel<<<(unsigned)((stot + 255) / 256), 256, 0, stream>>>(
      sA, outp + vtot, 625, stot);
}